// Nlplus_71330816852650
// MI455X (gfx1250) — compile-verified
//
#include <hip/hip_runtime.h>
#include <hip/hip_bf16.h>

typedef __attribute__((ext_vector_type(2))) float v2f;
typedef __attribute__((ext_vector_type(8))) float v8f;

#define L2E 1.4426950408889634f
#define NTHREADS 256

// One workgroup per row: single-pass online softmax (1 v_exp_f32 per element),
// b128 streaming loads, LDS (m,s) tree-combine, then the sparse-gradient algebra.
__global__ __launch_bounds__(NTHREADS) void nlp_row_kernel(
    const float* __restrict__ out, const int* __restrict__ tgt,
    const int* __restrict__ noff, float* __restrict__ ws, int C) {
  const int row = blockIdx.x;
  const int t   = threadIdx.x;
  const float* rp = out + (size_t)row * (size_t)C;

  float m = -3.0e38f;
  float s = 0.0f;

  // Vector path (b128) when rows stay 16B-aligned.
  const int nv = ((C & 3) == 0) ? (C >> 2) : 0;
  const float4* rv = (const float4*)rp;
  for (int i = t; i < nv; i += NTHREADS) {
    float4 v = rv[i];
    float mx = fmaxf(fmaxf(v.x, v.y), fmaxf(v.z, v.w));
    if (mx > m) {  // rare: rescale running sum to new max
      s *= __builtin_amdgcn_exp2f((m - mx) * L2E);
      m = mx;
    }
    s += __builtin_amdgcn_exp2f((v.x - m) * L2E)
       + __builtin_amdgcn_exp2f((v.y - m) * L2E)
       + __builtin_amdgcn_exp2f((v.z - m) * L2E)
       + __builtin_amdgcn_exp2f((v.w - m) * L2E);
  }
  // Scalar tail (also covers C % 4 != 0 entirely).
  for (int i = (nv << 2) + t; i < C; i += NTHREADS) {
    float x = rp[i];
    if (x > m) {
      s *= __builtin_amdgcn_exp2f((m - x) * L2E);
      m = x;
    }
    s += __builtin_amdgcn_exp2f((x - m) * L2E);
  }

  // Combine (m, s) pairs across the 256 threads.
  __shared__ float sm[NTHREADS];
  __shared__ float ssum[NTHREADS];
  sm[t] = m;
  ssum[t] = s;
  __syncthreads();
  for (int off = NTHREADS >> 1; off > 0; off >>= 1) {
    if (t < off) {
      float m2 = sm[t + off];
      float s2 = ssum[t + off];
      float M = fmaxf(m, m2);
      s = s  * __builtin_amdgcn_exp2f((m  - M) * L2E)
        + s2 * __builtin_amdgcn_exp2f((m2 - M) * L2E);
      m = M;
      sm[t] = m;
      ssum[t] = s;
    }
    __syncthreads();
  }

  if (t == 0) {
    int y = tgt[row];
    int k = (y + noff[row]) % C;  // target_neg
    float o_pos = rp[y];
    float o_neg = rp[k];
    float invS = 1.0f / s;
    float p_y = __builtin_amdgcn_exp2f((o_pos - m) * L2E) * invS;
    float p_k = __builtin_amdgcn_exp2f((o_neg - m) * L2E) * invS;
    p_y = fminf(fmaxf(p_y, 1e-7f), 1.0f);
    p_k = fminf(fmaxf(p_k, 1e-7f), 1.0f);
    float tt = 1.0f - (p_k - p_y);
    float g_neg = -(p_k * (p_y + p_k)) * tt - p_k * (1.0f - p_k) * tt;
    float g_pos = p_k * tt + p_k * p_y * tt;
    ws[row] = g_neg * o_neg + g_pos * o_pos;
  }
}

// Deterministic final reduction: 256-thread pre-sum, then wave 0 finishes the
// 256 partials with V_WMMA_F32_16X16X4_F32 (B = all-ones -> D rows hold
// 16-value sums, accumulated through C). f32-exact, no atomics.
__global__ __launch_bounds__(NTHREADS) void nlp_reduce_kernel(
    const float* __restrict__ ws, float* __restrict__ out, int n) {
  __shared__ float part[NTHREADS];
  __shared__ float fin[2];
  const int t = threadIdx.x;

  float s = 0.0f;
  for (int i = t; i < n; i += NTHREADS) s += ws[i];
  part[t] = s;
  __syncthreads();

  if (t < 32) {  // full wave, EXEC all ones inside this branch
    v2f b;
    b.x = 1.0f;
    b.y = 1.0f;
    v8f c = {};
    #pragma unroll
    for (int it = 0; it < 4; ++it) {
      v2f a;
      a.x = part[it * 64 + t];
      a.y = part[it * 64 + 32 + t];
      // D = A(16x4) * ones(4x16) + C : every D column holds the 16 row-sums.
      c = __builtin_amdgcn_wmma_f32_16x16x4_f32(
          /*neg_a=*/false, a, /*neg_b=*/false, b,
          /*c_mod=*/(short)0, c, /*reuse_a=*/false, /*reuse_b=*/false);
    }
    float l = c[0] + c[1] + c[2] + c[3] + c[4] + c[5] + c[6] + c[7];
    if (t == 0)  fin[0] = l;  // rows 0..7  (column 0)
    if (t == 16) fin[1] = l;  // rows 8..15 (column 0)
  }
  __syncthreads();

  if (t == 0) out[0] = -(fin[0] + fin[1]) / (float)n;
}

extern "C" void kernel_launch(void* const* d_in, const int* in_sizes, int n_in,
                              void* d_out, int out_size, void* d_ws, size_t ws_size,
                              hipStream_t stream) {
  const float* output = (const float*)d_in[0];
  const int*   target = (const int*)d_in[1];
  const int*   negoff = (const int*)d_in[2];
  float* out = (float*)d_out;
  float* ws  = (float*)d_ws;

  const int B = in_sizes[1];
  const int C = in_sizes[0] / B;

  nlp_row_kernel<<<B, NTHREADS, 0, stream>>>(output, target, negoff, ws, C);
  nlp_reduce_kernel<<<1, NTHREADS, 0, stream>>>(ws, out, B);
}